// FocalLoss_13365938226077
// MI455X (gfx1250) — compile-verified
//
#include <hip/hip_runtime.h>

typedef float v4f __attribute__((ext_vector_type(4)));
typedef int   v4i __attribute__((ext_vector_type(4)));
typedef float v2f __attribute__((ext_vector_type(2)));
typedef float v8f __attribute__((ext_vector_type(8)));

#define FL_ALPHA   0.9f
#define LOG10_OF_2 0.30102999566398119521f   // log10(x) = log2(x) * log10(2)

#if defined(__has_builtin)
#if __has_builtin(__builtin_amdgcn_wmma_f32_16x16x4_f32)
#define HAVE_WMMA_F32X4 1
#endif
#endif

__device__ __forceinline__ float focal_term(float p, int y) {
    // pt = y ? p : 1-p ; (1-pt) = y ? 1-p : p ; weight = y ? a : 1-a
    float pt  = y ? p        : 1.0f - p;
    float omp = y ? 1.0f - p : p;
    float w   = y ? FL_ALPHA : (1.0f - FL_ALPHA);
    float neg_log10 = -LOG10_OF_2 * __log2f(pt);   // v_log_f32 (trans)
    return w * (omp * omp) * neg_log10;
}

__global__ void focal_partial_kernel(const float* __restrict__ prob,
                                     const int*   __restrict__ targ,
                                     float* __restrict__ partials,
                                     long long n) {
    const long long n4     = n >> 2;
    const v4f* __restrict__ p4 = (const v4f*)prob;
    const v4i* __restrict__ t4 = (const v4i*)targ;

    const long long stride = (long long)gridDim.x * blockDim.x;
    long long i = (long long)blockIdx.x * blockDim.x + threadIdx.x;

    float acc = 0.0f;
    for (; i < n4; i += stride) {
        // single-use stream: NT hint keeps it out of the way of L2 reuse
        v4f pv = __builtin_nontemporal_load(&p4[i]);
        v4i tv = __builtin_nontemporal_load(&t4[i]);
        acc += focal_term(pv.x, tv.x);
        acc += focal_term(pv.y, tv.y);
        acc += focal_term(pv.z, tv.z);
        acc += focal_term(pv.w, tv.w);
    }
    // generic tail (n % 4) — empty for the reference shape
    if (blockIdx.x == 0 && threadIdx.x == 0) {
        for (long long k = (n4 << 2); k < n; ++k)
            acc += focal_term(prob[k], targ[k]);
    }

    // wave32 butterfly reduce
    for (int off = 16; off > 0; off >>= 1)
        acc += __shfl_xor(acc, off, 32);

    __shared__ float wsum[8];
    const int lane = threadIdx.x & 31;
    const int wave = threadIdx.x >> 5;
    if (lane == 0) wsum[wave] = acc;
    __syncthreads();

    if (wave == 0) {
        float v = (lane < (int)(blockDim.x >> 5)) ? wsum[lane] : 0.0f;
        for (int off = 16; off > 0; off >>= 1)
            v += __shfl_xor(v, off, 32);
        if (lane == 0) partials[blockIdx.x] = v;
    }
}

// Final deterministic reduction of `nparts` block partials, one full wave32.
// Uses V_WMMA_F32_16X16X4_F32 (f32 A/B, f32 accum -> no precision loss):
// with A = ones(16x4), D = A*B + C accumulates column sums of B, so each
// WMMA folds 64 partials into the accumulator; sum(D) == 16 * sum(partials).
__global__ void focal_final_kernel(const float* __restrict__ partials,
                                   float* __restrict__ out,
                                   int nparts, float inv_n) {
    const int lane = threadIdx.x;   // 32 threads, EXEC all ones (WMMA req.)

#if defined(HAVE_WMMA_F32X4)
    v2f a; a.x = 1.0f; a.y = 1.0f;  // A = 16x4 ones (2 VGPRs / lane)
    v8f c = {};                     // 16x16 f32 accumulator
    const int chunks = nparts >> 6; // 64 partials per WMMA
    for (int i = 0; i < chunks; ++i) {
        v2f b;
        b.x = partials[i * 64 + lane];
        b.y = partials[i * 64 + 32 + lane];
        c = __builtin_amdgcn_wmma_f32_16x16x4_f32(
                /*neg_a=*/false, a, /*neg_b=*/false, b,
                /*c_mod=*/(short)0, c, /*reuse_a=*/false, /*reuse_b=*/false);
    }
    float s8 = 0.0f;
    for (int k = 0; k < 8; ++k) s8 += c[k];
    float extra = 0.0f;
    for (int k = chunks * 64 + lane; k < nparts; k += 32)
        extra += partials[k];
    float comb = s8 * (1.0f / 16.0f) + extra;
#else
    float comb = 0.0f;
    for (int k = lane; k < nparts; k += 32)
        comb += partials[k];
#endif

    for (int off = 16; off > 0; off >>= 1)
        comb += __shfl_xor(comb, off, 32);
    if (lane == 0)
        out[0] = comb * inv_n;
}

extern "C" void kernel_launch(void* const* d_in, const int* in_sizes, int n_in,
                              void* d_out, int out_size, void* d_ws, size_t ws_size,
                              hipStream_t stream) {
    const float* prob = (const float*)d_in[0];
    const int*   targ = (const int*)d_in[1];
    float*       out  = (float*)d_out;
    float*       partials = (float*)d_ws;           // 1024 floats = 4 KB

    const long long n = (long long)in_sizes[0];
    const int NBLOCKS = 1024;                       // fixed -> deterministic
    const int NTHREADS = 256;                       // 8 wave32 per block

    focal_partial_kernel<<<NBLOCKS, NTHREADS, 0, stream>>>(prob, targ, partials, n);
    focal_final_kernel<<<1, 32, 0, stream>>>(partials, out, NBLOCKS,
                                             1.0f / (float)n);
}